// DistanceLoss_87900800680551
// MI455X (gfx1250) — compile-verified
//
#include <hip/hip_runtime.h>
#include <hip/hip_bf16.h>
#include <stdint.h>

#define WIN     5
#define THRESH  7.0f
#define BLK     256
#define HALO_L  4
#define HALO_R  5
#define TILE_N  (BLK + HALO_L + HALO_R)   // 265 d-values actually consumed
#define TILE_PAD 272                       // 68*4 floats, b128-chunk padded
#define NCHUNK  67                         // 67*4 = 268 >= 265 floats staged

// Deterministic block reduction: wave32 shuffle tree + LDS across 8 waves.
__device__ __forceinline__ float block_reduce(float v, float* red) {
    const int lane = threadIdx.x & 31;
    const int wid  = threadIdx.x >> 5;
#pragma unroll
    for (int o = 16; o > 0; o >>= 1) v += __shfl_down(v, o, 32);
    if (lane == 0) red[wid] = v;
    __syncthreads();
    float r = 0.f;
    if (wid == 0) {
        r = (lane < (int)(blockDim.x >> 5)) ? red[lane] : 0.f;
#pragma unroll
        for (int o = 16; o > 0; o >>= 1) r += __shfl_down(r, o, 32);
    }
    return r;  // valid in thread 0
}

// Kernel 1: gather triangle vertices, emit edge lengths d[3f..3f+2],
// accumulate hinge loss sum((7-d)^2 for d<7) as block partials.
// faces stream once from HBM (b96/lane); points (16 MB) stay L2-resident.
__global__ void edge_kernel(const float2* __restrict__ pts,
                            const int*    __restrict__ faces,
                            float*        __restrict__ d,
                            float*        __restrict__ partial,
                            int F) {
    __shared__ float red[8];
    int f = blockIdx.x * blockDim.x + threadIdx.x;
    float acc = 0.f;
    if (f < F) {
        int i0 = faces[3 * f + 0];
        int i1 = faces[3 * f + 1];
        int i2 = faces[3 * f + 2];
        float2 p0 = pts[i0], p1 = pts[i1], p2 = pts[i2];
        // edges = roll(tri,1,axis=1) - tri:
        //   d[3f+0]=|p2-p0|, d[3f+1]=|p0-p1|, d[3f+2]=|p1-p2|
        float dx, dy;
        dx = p2.x - p0.x; dy = p2.y - p0.y; float d0 = sqrtf(dx * dx + dy * dy);
        dx = p0.x - p1.x; dy = p0.y - p1.y; float d1 = sqrtf(dx * dx + dy * dy);
        dx = p1.x - p2.x; dy = p1.y - p2.y; float d2 = sqrtf(dx * dx + dy * dy);
        d[3 * f + 0] = d0;
        d[3 * f + 1] = d1;
        d[3 * f + 2] = d2;
        float t;
        t = THRESH - d0; t = t > 0.f ? t : 0.f; acc += t * t;
        t = THRESH - d1; t = t > 0.f ? t : 0.f; acc += t * t;
        t = THRESH - d2; t = t > 0.f ? t : 0.f; acc += t * t;
    }
    float s = block_reduce(acc, red);
    if (threadIdx.x == 0) partial[blockIdx.x] = s;
}

// Kernel 2: sliding-window mean-diff + exp. CDNA5 async global->LDS staging
// with b128 granules (tile global offset 1024*b-16 is 16B aligned for every
// block; block 0 clamps to 0 and shifts the LDS destination by 4 floats --
// tile[0..3] is never read there since the prev-window clamps to lo>=0).
// Tracked on ASYNCcnt; each wave waits on its own loads, barrier publishes.
__global__ void window_kernel(const float* __restrict__ d,
                              float*       __restrict__ partial,
                              int M, int Mw) {
    __shared__ __align__(16) float tile[TILE_PAD];
    __shared__ float red[8];
    const int j0 = blockIdx.x * BLK;

    int gbase = j0 - HALO_L;           // 16B-aligned global float index
    int lshift = 0;
    if (gbase < 0) { lshift = HALO_L; gbase = 0; }   // block 0 only

    const unsigned long long base = (unsigned long long)(uintptr_t)d;
    const int c = threadIdx.x;
    if (c < NCHUNK) {
        unsigned voff = (unsigned)(gbase + 4 * c) * 4u;          // 16B aligned
        unsigned lds  = (unsigned)(uintptr_t)(&tile[lshift + 4 * c]); // 16B aligned
        asm volatile("global_load_async_to_lds_b128 %0, %1, %2"
                     :: "v"(lds), "v"(voff), "s"(base) : "memory");
    }
    asm volatile("s_wait_asynccnt 0" ::: "memory");  // this wave's loads done
    __syncthreads();                                 // all waves' loads visible

    int j = j0 + threadIdx.x;
    float val = 0.f;
    if (j < Mw) {
        int b = threadIdx.x + HALO_L;                // tile index of d[j]
        // avg_next: always 5 elements d[j+1..j+5]
        float sN = tile[b + 1] + tile[b + 2] + tile[b + 3] + tile[b + 4] + tile[b + 5];
        // avg_prev: d[max(0,j-4)..j], 1..5 elements (only j<4 differs)
        int cnt = (j + 1 < WIN) ? (j + 1) : WIN;
        float sP = 0.f;
#pragma unroll
        for (int k = 0; k < WIN; ++k) sP += (k < cnt) ? tile[b - k] : 0.f;
        float diff = sN * 0.2f - sP / (float)cnt;
        val = __expf(fabsf(diff));
    }
    float s = block_reduce(val, red);
    if (threadIdx.x == 0) partial[blockIdx.x] = s;
}

// Kernel 3: deterministic final reduction of both partial arrays.
__global__ void final_kernel(const float* __restrict__ p1, int n1,
                             const float* __restrict__ p2, int n2,
                             float*       __restrict__ out) {
    __shared__ float red[8];
    float acc = 0.f;
    for (int i = threadIdx.x; i < n1; i += blockDim.x) acc += p1[i];
    for (int i = threadIdx.x; i < n2; i += blockDim.x) acc += p2[i];
    float s = block_reduce(acc, red);
    if (threadIdx.x == 0) out[0] = s;
}

extern "C" void kernel_launch(void* const* d_in, const int* in_sizes, int n_in,
                              void* d_out, int out_size, void* d_ws, size_t ws_size,
                              hipStream_t stream) {
    const float2* pts  = (const float2*)d_in[0];   // [N,2] f32
    const int*   faces = (const int*)d_in[1];      // [F,3] i32

    const int F  = in_sizes[1] / 3;
    const int M  = 3 * F;          // 12,000,000 edge lengths
    const int Mw = M - WIN;        // number of windows

    const int g1 = (F  + BLK - 1) / BLK;   // 15,625
    const int g2 = (Mw + BLK - 1) / BLK;   // 46,875

    float* ws = (float*)d_ws;
    float* d  = ws;                        // M floats (48 MB)
    float* p1 = ws + (size_t)M;            // g1 partials (hinge loss)
    float* p2 = p1 + g1;                   // g2 partials (window loss)

    edge_kernel  <<<g1, BLK, 0, stream>>>(pts, faces, d, p1, F);
    window_kernel<<<g2, BLK, 0, stream>>>(d, p2, M, Mw);
    final_kernel <<<1,  BLK, 0, stream>>>(p1, g1, p2, g2, (float*)d_out);
}